// BackwardDense_8950711846020
// MI455X (gfx1250) — compile-verified
//
#include <hip/hip_runtime.h>

// ---------------------------------------------------------------------------
// BackwardDense (linear branch), MI455X / gfx1250, wave32 + WMMA f32 + TDM.
//
//   w_out_*' [b,i,n] = sum_o kernel[i,o] * w_out_*[b,0,o,n]   (pos+neg == w exactly)
//   b_out_*' [b,n]   = b_out_*[b,n] + sum_o bias[o] * w_out_*[b,0,o,n]
//
// B=32, DIN=64, DOUT=512, N=2048, all float32.
// HBM floor ~303 MB -> ~13 us @ 23.3 TB/s. 4.3 G MACs in full f32 via
// v_wmma_f32_16x16x4_f32. W stream staged to LDS by the Tensor Data Mover
// (tensor_load_to_lds + s_wait_tensorcnt), D# pad fields give a 136-dw padded
// block stride for conflict-free B-fragment reads. Each wave owns TWO N-tiles
// so every 4 A-fragment loads feed 8 WMMAs.
// ---------------------------------------------------------------------------

typedef float v2f __attribute__((ext_vector_type(2)));
typedef float v4f __attribute__((ext_vector_type(4)));
typedef float v8f __attribute__((ext_vector_type(8)));
typedef unsigned int u32x4 __attribute__((ext_vector_type(4)));
typedef int i32x4 __attribute__((ext_vector_type(4)));
typedef int i32x8 __attribute__((ext_vector_type(8)));

#define DIN    64
#define DOUT   512
#define NN     2048
#define NBATCH 32
#define KC     32    // DOUT-chunk staged in LDS per step
#define NSLAB  256   // N columns per workgroup (8 waves x 2 tiles x 16)
#define SBLK   136   // padded 128-dword block stride (TDM pad: +8dw/128dw)
#define LSTR   (2 * SBLK)  // floats per staged LDS row (272)

#if __has_builtin(__builtin_amdgcn_tensor_load_to_lds)
#define USE_TDM 1
#else
#define USE_TDM 0
#endif

#if __has_builtin(__builtin_amdgcn_s_wait_tensorcnt)
#define WAIT_TENSORCNT0() __builtin_amdgcn_s_wait_tensorcnt(0)
#else
#define WAIT_TENSORCNT0() asm volatile("s_wait_tensorcnt 0x0" ::: "memory")
#endif

#if USE_TDM
// One TDM descriptor: load a [KC x NSLAB] f32 tile (row stride NN floats) from
// gsrc into LDS at lds_byte_off. pad_enable inserts 8 dwords after every 128
// dwords moved, so rows land with the [KC][LSTR] padded layout.
__device__ __forceinline__ void tdm_load_chunk(const float* gsrc,
                                               unsigned lds_byte_off)
{
    unsigned long long ga = (unsigned long long)(size_t)gsrc;

    // D# group 0: count=1 | lds_addr | global_addr[56:0] | type=2
    u32x4 g0;
    g0[0] = 1u;
    g0[1] = lds_byte_off;
    g0[2] = (unsigned)ga;
    g0[3] = (unsigned)((ga >> 32) & 0x1FFFFFFu) | (2u << 30);

    // D# group 1: data_size=4B(2), pad_enable, pad_interval=128dw(6),
    // pad_amount=8dw(7); tensor 2048x512, tile 256x32, dim0 stride 2048.
    i32x8 g1;
    g1[0] = (2 << 16) | (1 << 20) | (6 << 22) | (7 << 25);
    g1[1] = (int)((unsigned)NN << 16);      // tensor_dim0[15:0] in [31:16]
    g1[2] = (int)((unsigned)DOUT << 16);    // tensor_dim0[31:16]=0 | tensor_dim1[15:0]
    g1[3] = (int)((unsigned)NSLAB << 16);   // tensor_dim1[31:16]=0 | tile_dim0=256
    g1[4] = KC;                             // tile_dim1=32, tile_dim2=0
    g1[5] = NN;                             // tensor_dim0_stride low32
    g1[6] = 0;
    g1[7] = 0;

    i32x4 z4 = {0, 0, 0, 0};
#if defined(__clang_major__) && (__clang_major__ >= 23)
    i32x8 z8 = {0, 0, 0, 0, 0, 0, 0, 0};
    __builtin_amdgcn_tensor_load_to_lds(g0, g1, z4, z4, z8, 0);
#else
    __builtin_amdgcn_tensor_load_to_lds(g0, g1, z4, z4, 0);
#endif
}
#endif // USE_TDM

__global__ __launch_bounds__(256)
void bwd_dense_wmma(const float* __restrict__ kmat,   // [DIN, DOUT]
                    const float* __restrict__ bias,   // [DOUT]
                    const float* __restrict__ w_u,    // [B, DOUT, NN]
                    const float* __restrict__ b_u,    // [B, NN]
                    const float* __restrict__ w_l,
                    const float* __restrict__ b_l,
                    float* __restrict__ out_wu,       // [B, DIN, NN]
                    float* __restrict__ out_bu,       // [B, NN]
                    float* __restrict__ out_wl,
                    float* __restrict__ out_bl)
{
    __shared__ __align__(16) float sW[2][KC][LSTR];   // double-buffered chunk (~68 KB)
    __shared__ __align__(16) float sBias[DOUT];

    const int nslab = blockIdx.x * NSLAB;
    const int batch = blockIdx.y;
    const bool isL  = (blockIdx.z != 0);

    const float* __restrict__ W   = (isL ? w_l : w_u) + (size_t)batch * DOUT * NN;
    const float* __restrict__ Bin = (isL ? b_l : b_u) + (size_t)batch * NN;
    float* __restrict__ Cout = (isL ? out_wl : out_wu) + (size_t)batch * DIN * NN;
    float* __restrict__ Bout = (isL ? out_bl : out_bu) + (size_t)batch * NN;

    const int tid  = threadIdx.x;
    const int wave = tid >> 5;     // 0..7
    const int lane = tid & 31;
    const int half = lane >> 4;    // 0/1 -> K-pair select in A/B fragments
    const int l15  = lane & 15;

    // Stage bias into LDS once.
    sBias[tid]       = bias[tid];
    sBias[tid + 256] = bias[tid + 256];

#if !USE_TDM
    // Fallback manual loader: 8 x (256 threads x float4) per chunk.
    const int lr = tid >> 6;               // 0..3 base row
    const int lc = (tid & 63) * 4;         // 0..252
    const int sc = (lc & 127) + (lc >> 7) * SBLK;  // padded LDS column
    auto load_chunk = [&](int buf, int k0) {
        #pragma unroll
        for (int rr = 0; rr < KC; rr += 4) {
            v4f v = *(const v4f*)(W + (size_t)(k0 + lr + rr) * NN + nslab + lc);
            *(v4f*)(&sW[buf][lr + rr][sc]) = v;
        }
    };
#endif

    // Prime first chunk.
#if USE_TDM
    if (wave == 0) {
        tdm_load_chunk(W + nslab, (unsigned)(size_t)&sW[0][0][0]);
        WAIT_TENSORCNT0();
    }
#else
    load_chunk(0, 0);
#endif

    // Accumulators: 4 M-tiles (DIN=64) x 2 N-tiles per wave.
    v8f acc00 = {}, acc10 = {}, acc20 = {}, acc30 = {};   // N-tile 0
    v8f acc01 = {}, acc11 = {}, acc21 = {}, acc31 = {};   // N-tile 1
    float biasAcc0 = 0.0f, biasAcc1 = 0.0f;

    const int col = wave * 16 + l15;       // N-tile 0 column within slab
    // N-tile 1 column = col + 128 (LDS padded offset: +SBLK)
    // A-fragment base: lane l15 = M row, half-wave selects K pair {0,1}/{2,3}.
    const float* Abase = kmat + (size_t)l15 * DOUT + half * 2;

    __syncthreads();

    for (int kc = 0; kc < DOUT / KC; ++kc) {
        const int buf = kc & 1;
        const int k0  = kc * KC;

        // Kick off DMA of the next chunk into the other buffer.
        if (kc + 1 < DOUT / KC) {
#if USE_TDM
            if (wave == 0)
                tdm_load_chunk(W + (size_t)(k0 + KC) * NN + nslab,
                               (unsigned)(size_t)&sW[buf ^ 1][0][0]);
#else
            load_chunk(buf ^ 1, k0 + KC);
#endif
        }

        // --- WMMA core: 4 A-loads feed 8 v_wmma per K-step of 4 ---
        #pragma unroll
        for (int kk = 0; kk < KC; kk += 4) {
            const int r0 = kk + half * 2;
            // B fragments (4x16): VGPR0 = rows {kk, kk+2}, VGPR1 = {kk+1, kk+3}
            v2f bf0, bf1;
            bf0.x = sW[buf][r0 + 0][col];
            bf0.y = sW[buf][r0 + 1][col];
            bf1.x = sW[buf][r0 + 0][col + SBLK];
            bf1.y = sW[buf][r0 + 1][col + SBLK];

            // Fused bias reduction: B fragments visit every (row,col) of the
            // chunk exactly once across the two half-waves.
            v2f bb = *(const v2f*)(&sBias[k0 + r0]);
            biasAcc0 = fmaf(bb.x, bf0.x, fmaf(bb.y, bf0.y, biasAcc0));
            biasAcc1 = fmaf(bb.x, bf1.x, fmaf(bb.y, bf1.y, biasAcc1));

            const int kg = k0 + kk;
            v2f a0 = *(const v2f*)(Abase + (size_t)(0 * 16) * DOUT + kg);
            v2f a1 = *(const v2f*)(Abase + (size_t)(1 * 16) * DOUT + kg);
            v2f a2 = *(const v2f*)(Abase + (size_t)(2 * 16) * DOUT + kg);
            v2f a3 = *(const v2f*)(Abase + (size_t)(3 * 16) * DOUT + kg);

            acc00 = __builtin_amdgcn_wmma_f32_16x16x4_f32(false, a0, false, bf0,
                                                          (short)0, acc00, false, false);
            acc10 = __builtin_amdgcn_wmma_f32_16x16x4_f32(false, a1, false, bf0,
                                                          (short)0, acc10, false, false);
            acc20 = __builtin_amdgcn_wmma_f32_16x16x4_f32(false, a2, false, bf0,
                                                          (short)0, acc20, false, false);
            acc30 = __builtin_amdgcn_wmma_f32_16x16x4_f32(false, a3, false, bf0,
                                                          (short)0, acc30, false, false);
            acc01 = __builtin_amdgcn_wmma_f32_16x16x4_f32(false, a0, false, bf1,
                                                          (short)0, acc01, false, false);
            acc11 = __builtin_amdgcn_wmma_f32_16x16x4_f32(false, a1, false, bf1,
                                                          (short)0, acc11, false, false);
            acc21 = __builtin_amdgcn_wmma_f32_16x16x4_f32(false, a2, false, bf1,
                                                          (short)0, acc21, false, false);
            acc31 = __builtin_amdgcn_wmma_f32_16x16x4_f32(false, a3, false, bf1,
                                                          (short)0, acc31, false, false);
        }

#if USE_TDM
        if (wave == 0) WAIT_TENSORCNT0();   // next chunk landed in LDS
#endif
        __syncthreads();                    // publish next buf / retire this buf
    }

    // --- bias output: combine half-wave partials (wave32 xor-16 shuffle) ---
    const float bTot0 = biasAcc0 + __shfl_xor(biasAcc0, 16, 32);
    const float bTot1 = biasAcc1 + __shfl_xor(biasAcc1, 16, 32);
    if (half == 0) {
        const int n = nslab + col;
        Bout[n]       = Bin[n]       + bTot0;
        Bout[n + 128] = Bin[n + 128] + bTot1;
    }

    // --- C output: v8f layout -> row = mt*16 + half*8 + j ---
    #pragma unroll
    for (int j = 0; j < 8; ++j) {
        const int r = half * 8 + j;
        float* c0 = Cout + (size_t)r * NN + nslab + col;
        c0[(size_t)( 0) * NN]       = acc00[j];
        c0[(size_t)(16) * NN]       = acc10[j];
        c0[(size_t)(32) * NN]       = acc20[j];
        c0[(size_t)(48) * NN]       = acc30[j];
        c0[(size_t)( 0) * NN + 128] = acc01[j];
        c0[(size_t)(16) * NN + 128] = acc11[j];
        c0[(size_t)(32) * NN + 128] = acc21[j];
        c0[(size_t)(48) * NN + 128] = acc31[j];
    }
}

extern "C" void kernel_launch(void* const* d_in, const int* in_sizes, int n_in,
                              void* d_out, int out_size, void* d_ws, size_t ws_size,
                              hipStream_t stream) {
    const float* kmat = (const float*)d_in[0];   // [64, 512]
    const float* bias = (const float*)d_in[1];   // [512]
    const float* w_u  = (const float*)d_in[2];   // [32, 1, 512, 2048]
    const float* b_u  = (const float*)d_in[3];   // [32, 1, 2048]
    const float* w_l  = (const float*)d_in[4];
    const float* b_l  = (const float*)d_in[5];

    float* out = (float*)d_out;
    // Tuple outputs concatenated flat in return order:
    float* out_wu = out;                                   // 32*64*2048 = 4194304
    float* out_bu = out + 4194304;                         // 32*2048    = 65536
    float* out_wl = out + 4194304 + 65536;                 // 4194304
    float* out_bl = out + 4194304 + 65536 + 4194304;       // 65536

    dim3 grid(NN / NSLAB, NBATCH, 2);   // (8, 32, 2) = 512 workgroups
    dim3 block(256);                    // 8 waves (wave32)
    bwd_dense_wmma<<<grid, block, 0, stream>>>(kmat, bias, w_u, b_u, w_l, b_l,
                                               out_wu, out_bu, out_wl, out_bl);
}